// Net_36885179138053
// MI455X (gfx1250) — compile-verified
//
#include <hip/hip_runtime.h>
#include <hip/hip_bf16.h>

typedef __attribute__((ext_vector_type(2))) float v2f;
typedef __attribute__((ext_vector_type(8))) float v8f;

#define N_NODES 100000
#define N_EDGES 1600000
#define HIDDEN 64
#define N_GRAPHS 512
#define EPS_MSG 1e-7f
#define BN_EPS 1e-5f

// ---------------- input projection: hs = x@Wsrc+bs ; hd = x@Wdst+bd ----------
__global__ void in_proj_kernel(const float* __restrict__ x,
                               const float* __restrict__ ws, const float* __restrict__ bs,
                               const float* __restrict__ wd, const float* __restrict__ bd,
                               float* __restrict__ hs, float* __restrict__ hd) {
    int gid = blockIdx.x * blockDim.x + threadIdx.x;
    if (gid >= N_NODES * HIDDEN) return;
    int n = gid >> 6, c = gid & 63;
    float as = bs[c], ad = bd[c];
#pragma unroll
    for (int k = 0; k < 9; ++k) {
        float xv = x[n * 9 + k];
        as += xv * ws[k * HIDDEN + c];
        ad += xv * wd[k * HIDDEN + c];
    }
    hs[gid] = as;
    hd[gid] = ad;
}

// ---------------- edge pass 1: per-(dst,ch) max of msg (msg > 0 -> int-bit max ok)
__global__ void edge_max_kernel(const float* __restrict__ hsrc,
                                const int* __restrict__ src, const int* __restrict__ dst,
                                float* __restrict__ mx) {
    int gid = blockIdx.x * blockDim.x + threadIdx.x;
    if (gid >= N_EDGES * 16) return;
    int e = gid >> 4, c4 = (gid & 15) << 2;
    int sn = src[e], dn = dst[e];
    const float4 h = *(const float4*)(hsrc + (size_t)sn * HIDDEN + c4);
    int* m = (int*)(mx + (size_t)dn * HIDDEN + c4);
    atomicMax(&m[0], __float_as_int(fmaxf(h.x, 0.f) + EPS_MSG));
    atomicMax(&m[1], __float_as_int(fmaxf(h.y, 0.f) + EPS_MSG));
    atomicMax(&m[2], __float_as_int(fmaxf(h.z, 0.f) + EPS_MSG));
    atomicMax(&m[3], __float_as_int(fmaxf(h.w, 0.f) + EPS_MSG));
}

// ---------------- edge pass 2: s += exp(m-mx) ; t += exp(m-mx)*m -------------
__global__ void edge_sum_kernel(const float* __restrict__ hsrc,
                                const int* __restrict__ src, const int* __restrict__ dst,
                                const float* __restrict__ mx,
                                float* __restrict__ sb, float* __restrict__ tb) {
    int gid = blockIdx.x * blockDim.x + threadIdx.x;
    if (gid >= N_EDGES * 16) return;
    int e = gid >> 4, c4 = (gid & 15) << 2;
    int sn = src[e], dn = dst[e];
    const float4 h = *(const float4*)(hsrc + (size_t)sn * HIDDEN + c4);
    const float4 mv = *(const float4*)(mx + (size_t)dn * HIDDEN + c4);
    float* sp = sb + (size_t)dn * HIDDEN + c4;
    float* tp = tb + (size_t)dn * HIDDEN + c4;
    float m0 = fmaxf(h.x, 0.f) + EPS_MSG, m1 = fmaxf(h.y, 0.f) + EPS_MSG;
    float m2 = fmaxf(h.z, 0.f) + EPS_MSG, m3 = fmaxf(h.w, 0.f) + EPS_MSG;
    float e0 = __expf(m0 - mv.x), e1 = __expf(m1 - mv.y);
    float e2 = __expf(m2 - mv.z), e3 = __expf(m3 - mv.w);
    atomicAdd(&sp[0], e0); atomicAdd(&tp[0], e0 * m0);
    atomicAdd(&sp[1], e1); atomicAdd(&tp[1], e1 * m1);
    atomicAdd(&sp[2], e2); atomicAdd(&tp[2], e2 * m2);
    atomicAdd(&sp[3], e3); atomicAdd(&tp[3], e3 * m3);
}

// ---------------- y = t/(s+1e-16) + hdst -------------------------------------
__global__ void combine_kernel(const float* __restrict__ tb, const float* __restrict__ sb,
                               const float* __restrict__ hdst, float* __restrict__ y) {
    int gid = blockIdx.x * blockDim.x + threadIdx.x;
    if (gid >= N_NODES * HIDDEN) return;
    y[gid] = tb[gid] / (sb[gid] + 1e-16f) + hdst[gid];
}

// ---------------- WMMA f32 GEMM: out[M,NC] = f(A)[M,K] @ W[K,NC] + bias ------
// Compile-time shapes; one wave owns a 16x16 tile, K/4 chained
// V_WMMA_F32_16X16X4_F32. W is staged transposed in LDS (shared by all 8
// waves of the block) so the B-operand is a single ds_load_b64 with an
// immediate offset; A-operand is a global_load_b64 with immediate offsets.
// BN variant folds the BatchNorm affine + ReLU into the A-side load.
template <int K, int NC, bool BN, bool RELU>
__global__ void gemm_wmma_kernel(const float* __restrict__ A, const float* __restrict__ W,
                                 const float* __restrict__ bias, float* __restrict__ out,
                                 const float* __restrict__ scale,
                                 const float* __restrict__ shift) {
    __shared__ float Wt[NC * K];                 // transposed: Wt[col*K + k]
    __shared__ float ssc[BN ? K : 1];
    __shared__ float ssh[BN ? K : 1];
    for (int i = threadIdx.x; i < K * NC; i += blockDim.x) {
        int k = i / NC, c = i - k * NC;          // coalesced row-major read
        Wt[c * K + k] = W[i];
    }
    if (BN) {
        for (int i = threadIdx.x; i < K; i += blockDim.x) {
            ssc[i] = scale[i];
            ssh[i] = shift[i];
        }
    }
    __syncthreads();

    const int lane = threadIdx.x & 31;
    const int wave = threadIdx.x >> 5;
    const int tile = blockIdx.x * (blockDim.x >> 5) + wave;
    const int mt = tile / (NC >> 4);
    const int nt = tile - mt * (NC >> 4);
    if (mt * 16 >= N_NODES) return;              // wave-uniform (never taken: M%16==0)

    const int half = lane >> 4;                  // 0: lanes 0-15, 1: lanes 16-31
    const int l = lane & 15;
    const int bcol = nt * 16 + l;
    const float* ap = A + (size_t)(mt * 16 + l) * K + (half << 1);
    const float* bp = Wt + (size_t)bcol * K + (half << 1);

    v8f acc = {};
#pragma unroll
    for (int kk = 0; kk < (K >> 2); ++kk) {
        float a0 = ap[kk * 4 + 0];
        float a1 = ap[kk * 4 + 1];
        if (BN) {
            const int k0 = kk * 4 + (half << 1);
            a0 = fmaxf(a0 * ssc[k0] + ssh[k0], 0.0f);
            a1 = fmaxf(a1 * ssc[k0 + 1] + ssh[k0 + 1], 0.0f);
        }
        v2f a, b;
        a.x = a0; a.y = a1;
        b.x = bp[kk * 4 + 0];
        b.y = bp[kk * 4 + 1];
        acc = __builtin_amdgcn_wmma_f32_16x16x4_f32(
            /*neg_a=*/false, a, /*neg_b=*/false, b,
            /*c_mod=*/(short)0, acc, /*reuse_a=*/false, /*reuse_b=*/false);
    }

    const float bv = bias[bcol];
    float* op = out + (size_t)(mt * 16 + half * 8) * NC + bcol;
#pragma unroll
    for (int r = 0; r < 8; ++r) {                // C/D: M = r + 8*half
        float v = acc[r] + bv;
        if (RELU) v = fmaxf(v, 0.0f);
        op[(size_t)r * NC] = v;
    }
}

// ---------------- BatchNorm statistics over N_NODES rows of 128 channels -----
__global__ void bn_stats_kernel(const float* __restrict__ h1,
                                float* __restrict__ sum, float* __restrict__ sumsq) {
    int c = threadIdx.x; // 128 threads = one channel each
    float s = 0.f, s2 = 0.f;
    for (int n = blockIdx.x; n < N_NODES; n += gridDim.x) {
        float v = h1[(size_t)n * 128 + c];
        s += v; s2 += v * v;
    }
    atomicAdd(&sum[c], s);
    atomicAdd(&sumsq[c], s2);
}

__global__ void bn_final_kernel(const float* __restrict__ sum, const float* __restrict__ sumsq,
                                const float* __restrict__ g, const float* __restrict__ be,
                                float* __restrict__ scale, float* __restrict__ shift) {
    int c = threadIdx.x;
    float inv_n = 1.0f / (float)N_NODES;
    float mu = sum[c] * inv_n;
    float var = sumsq[c] * inv_n - mu * mu;
    float sc = g[c] * rsqrtf(var + BN_EPS);
    scale[c] = sc;
    shift[c] = be[c] - mu * sc;
}

// ---------------- global mean pool + dense + sigmoid -------------------------
__global__ void pool_kernel(const float* __restrict__ h, const int* __restrict__ batch,
                            float* __restrict__ pool, float* __restrict__ cnt) {
    int gid = blockIdx.x * blockDim.x + threadIdx.x;
    if (gid >= N_NODES * HIDDEN) return;
    int n = gid >> 6, c = gid & 63;
    int g = batch[n];
    atomicAdd(&pool[(size_t)g * HIDDEN + c], h[gid]);
    if (c == 0) atomicAdd(&cnt[g], 1.0f);
}

__global__ void dense_sigmoid_kernel(const float* __restrict__ pool, const float* __restrict__ cnt,
                                     const float* __restrict__ wd, const float* __restrict__ bd,
                                     float* __restrict__ out) {
    __shared__ float red[HIDDEN];
    int g = blockIdx.x, c = threadIdx.x;
    float cv = fmaxf(cnt[g], 1.0f);
    red[c] = (pool[(size_t)g * HIDDEN + c] / cv) * wd[c];
    __syncthreads();
    for (int st = 32; st > 0; st >>= 1) {
        if (c < st) red[c] += red[c + st];
        __syncthreads();
    }
    if (c == 0) out[g] = 1.0f / (1.0f + __expf(-(red[0] + bd[0])));
}

extern "C" void kernel_launch(void* const* d_in, const int* in_sizes, int n_in,
                              void* d_out, int out_size, void* d_ws, size_t ws_size,
                              hipStream_t stream) {
    (void)in_sizes; (void)n_in; (void)out_size; (void)ws_size;
    const float* x       = (const float*)d_in[0];
    const int*   eidx    = (const int*)d_in[1];
    const int*   batch   = (const int*)d_in[2];
    const float* w_src   = (const float*)d_in[3];
    const float* b_src   = (const float*)d_in[4];
    const float* w_dst   = (const float*)d_in[5];
    const float* b_dst   = (const float*)d_in[6];
    const float* w_dense = (const float*)d_in[25];
    const float* b_dense = (const float*)d_in[26];
    float* out = (float*)d_out;

    const int* src = eidx;
    const int* dst = eidx + N_EDGES;

    // workspace layout (floats); S = N_NODES*HIDDEN
    const size_t S = (size_t)N_NODES * HIDDEN;
    float* wsf   = (float*)d_ws;
    float* hsrc0 = wsf + 0 * S;
    float* hdst0 = wsf + 1 * S;
    float* mx    = wsf + 2 * S;
    float* sb    = wsf + 3 * S;
    float* tb    = wsf + 4 * S;
    float* yb    = wsf + 5 * S;
    float* h1    = wsf + 6 * S;          // N_NODES*128 = 2*S
    float* hA    = wsf + 8 * S;
    float* hB    = wsf + 9 * S;
    float* bnsum   = wsf + 10 * S;
    float* bnsumsq = bnsum + 128;
    float* bnscale = bnsum + 256;
    float* bnshift = bnsum + 384;
    float* pool    = bnsum + 512;        // 512*64
    float* cnt     = pool + (size_t)N_GRAPHS * HIDDEN;

    const int n64 = N_NODES * HIDDEN;
    const int n64_blocks = (n64 + 255) / 256;
    const int e16_blocks = (N_EDGES * 16 + 255) / 256;

    // input projections
    in_proj_kernel<<<n64_blocks, 256, 0, stream>>>(x, w_src, b_src, w_dst, b_dst, hsrc0, hdst0);

    auto run_conv = [&](const float* hsrc, const float* hdst,
                        const float* w1, const float* b1, const float* g, const float* be,
                        const float* w2, const float* b2, float* hout) {
        hipMemsetAsync(mx, 0, S * sizeof(float), stream);   // msgs > 0, so 0.0f is a safe floor
        hipMemsetAsync(sb, 0, S * sizeof(float), stream);
        hipMemsetAsync(tb, 0, S * sizeof(float), stream);
        hipMemsetAsync(bnsum, 0, 256 * sizeof(float), stream);
        edge_max_kernel<<<e16_blocks, 256, 0, stream>>>(hsrc, src, dst, mx);
        edge_sum_kernel<<<e16_blocks, 256, 0, stream>>>(hsrc, src, dst, mx, sb, tb);
        combine_kernel<<<n64_blocks, 256, 0, stream>>>(tb, sb, hdst, yb);
        // h1 = y @ w1 + b1 : M=100000, K=64, NC=128 -> 50000 tiles, 8 waves/block
        gemm_wmma_kernel<64, 128, false, false>
            <<<(N_NODES / 16) * (128 / 16) / 8, 256, 0, stream>>>(
                yb, w1, b1, h1, nullptr, nullptr);
        bn_stats_kernel<<<512, 128, 0, stream>>>(h1, bnsum, bnsumsq);
        bn_final_kernel<<<1, 128, 0, stream>>>(bnsum, bnsumsq, g, be, bnscale, bnshift);
        // hout = relu( relu(BN(h1)) @ w2 + b2 ) : M=100000, K=128, NC=64
        gemm_wmma_kernel<128, 64, true, true>
            <<<(N_NODES / 16) * (64 / 16) / 8, 256, 0, stream>>>(
                h1, w2, b2, hout, bnscale, bnshift);
    };

    // conv 1 (distinct src/dst inputs)
    run_conv(hsrc0, hdst0,
             (const float*)d_in[7], (const float*)d_in[8], (const float*)d_in[9],
             (const float*)d_in[10], (const float*)d_in[11], (const float*)d_in[12], hA);
    // conv 2
    run_conv(hA, hA,
             (const float*)d_in[13], (const float*)d_in[14], (const float*)d_in[15],
             (const float*)d_in[16], (const float*)d_in[17], (const float*)d_in[18], hB);
    // conv 3
    run_conv(hB, hB,
             (const float*)d_in[19], (const float*)d_in[20], (const float*)d_in[21],
             (const float*)d_in[22], (const float*)d_in[23], (const float*)d_in[24], hA);

    // global mean pool + dense + sigmoid
    hipMemsetAsync(pool, 0, ((size_t)N_GRAPHS * HIDDEN + N_GRAPHS) * sizeof(float), stream);
    pool_kernel<<<n64_blocks, 256, 0, stream>>>(hA, batch, pool, cnt);
    dense_sigmoid_kernel<<<N_GRAPHS, HIDDEN, 0, stream>>>(pool, cnt, w_dense, b_dense, out);
}